// NextTaskGAT_197568496150
// MI455X (gfx1250) — compile-verified
//
#include <hip/hip_runtime.h>
#include <math.h>

#define N_NODES 50000
#define N_EDGES 800000
#define FDIM    256
#define NHEAD   4
#define CH      64
#define OUTD    64
#define NEG_SLOPE 0.2f

#define KC      64          // K-chunk staged in LDS per iteration
#define ASTRIDE 68          // 64 + 4 pad: stride%64 == 4 -> conflict-free b64 reads
#define BSTRIDE 260         // 256 + 4 pad: stride%64 == 4 -> conflict-free b64 reads

typedef float v2f __attribute__((ext_vector_type(2)));
typedef float v8f __attribute__((ext_vector_type(8)));

#if defined(__HIP_DEVICE_COMPILE__) && __has_builtin(__builtin_amdgcn_tensor_load_to_lds)
#define HAVE_TDM 1
#else
#define HAVE_TDM 0
#endif

#if HAVE_TDM
typedef unsigned u32x4 __attribute__((ext_vector_type(4)));
typedef int      i32x4 __attribute__((ext_vector_type(4)));
typedef int      i32x8 __attribute__((ext_vector_type(8)));

// LDS byte offset of a __shared__ object (generic -> AS(3) strips the aperture)
__device__ __forceinline__ unsigned lds_addr_of(const void* p) {
  return (unsigned)(unsigned long long)
         (__attribute__((address_space(3))) const void*)p;
}

// Issue one 2D TDM tile load: tile0 x tile1 elements of f32, source row
// stride = stride0 elements, dest = LDS rows padded 64->68 DWORDs
// (pad_interval code 5 = every 64 DWORDs, pad_amount code 3 = 4 DWORDs),
// rows beyond td1 are hardware zero-filled.  ISA 8.3/8.4 D# packing.
__device__ __forceinline__ void tdm_load_tile_f32(
    unsigned lds_byte_addr, const float* gsrc,
    unsigned tile0, unsigned tile1,
    unsigned td0, unsigned td1, unsigned stride0) {
  unsigned long long ga = (unsigned long long)gsrc;
  u32x4 g0;
  g0.x = 1u;                                                  // count=1, user
  g0.y = lds_byte_addr;                                       // lds_addr
  g0.z = (unsigned)ga;                                        // global[31:0]
  g0.w = (unsigned)((ga >> 32) & 0x01FFFFFFull) | (2u << 30); // global[56:32], type=2
  i32x8 g1;
  g1[0] = (int)((2u << 16)        // data_size = 4B
              | (1u << 20)        // pad_enable
              | (5u << 22)        // pad_interval: 64 DWORDs
              | (3u << 25));      // pad_amount:   4 DWORDs
  g1[1] = (int)((td0 & 0xFFFFu) << 16);                         // dim0[15:0]
  g1[2] = (int)(((td0 >> 16) & 0xFFFFu) | ((td1 & 0xFFFFu) << 16));
  g1[3] = (int)(((td1 >> 16) & 0xFFFFu) | ((tile0 & 0xFFFFu) << 16));
  g1[4] = (int)(tile1 & 0xFFFFu);                               // tile1, tile2=0
  g1[5] = (int)stride0;                                         // dim0_stride
  g1[6] = 0;
  g1[7] = 0;
  i32x4 z4 = {};
#if __clang_major__ >= 23
  i32x8 z8 = {};
  __builtin_amdgcn_tensor_load_to_lds(g0, g1, z4, z4, z8, 0);
#else
  __builtin_amdgcn_tensor_load_to_lds(g0, g1, z4, z4, 0);
#endif
}
#endif  // HAVE_TDM

// ---------------- order-preserving float <-> u32 key (for atomicMax) --------
__device__ __forceinline__ unsigned fkey(float f) {
  unsigned u = __float_as_uint(f);
  return (u & 0x80000000u) ? ~u : (u | 0x80000000u);
}
__device__ __forceinline__ float funkey(unsigned k) {
  unsigned u = (k & 0x80000000u) ? (k & 0x7FFFFFFFu) : ~k;
  return __uint_as_float(u);
}

// ---------------- fp32 WMMA GEMM: C[M,Nc] = A[M,K] @ B[K,Nc] (+bias) --------
// Block = 8 waves -> 128 rows x 32 cols.  Wave -> 16-row stripe, two 16x16
// col tiles sharing the A fragment.  B panel (K x 32, transposed) staged in
// LDS once by VALU; A staged in 64-wide K chunks by the Tensor Data Mover,
// double-buffered so the next DMA overlaps the current 32 WMMAs.  All LDS
// reads are ds_load_b64 with immediate offsets and pad-strides hitting all
// 64 banks exactly once.  WMMA runs with EXEC all-ones; only the final
// store is wave-uniformly predicated on the row tail.
__global__ __launch_bounds__(256) void gemm_wmma_f32(
    const float* __restrict__ A, const float* __restrict__ B,
    float* __restrict__ C, const float* __restrict__ bias,
    int M, int K, int Nc) {
  __shared__ __align__(16) float Bl[32 * BSTRIDE];       // 33,280 B
  __shared__ __align__(16) float Al[2][128 * ASTRIDE];   // 2 x 34,816 B

  const int tid  = threadIdx.x;
  const int lane = tid & 31;
  const int wave = tid >> 5;
  const int half = lane >> 4;           // K-half select per ISA 16x4 f32 layout
  const int m    = lane & 15;
  const int rowBase = blockIdx.x * 128;
  const int row0    = rowBase + wave * 16;
  const int col0    = blockIdx.y * 32;

  // stage B panel transposed: Bl[c][k] = B[k][col0+c]  (coalesced global reads)
  for (int idx = tid; idx < K * 32; idx += 256) {
    int k = idx >> 5, c = idx & 31;
    Bl[c * BSTRIDE + k] = B[(size_t)k * Nc + col0 + c];
  }

#if HAVE_TDM
  if (wave == 0)                        // one DMA per block; EXEC-independent
    tdm_load_tile_f32(lds_addr_of(&Al[0][0]), A + (size_t)rowBase * K,
                      KC, 128, (unsigned)K, (unsigned)(M - rowBase),
                      (unsigned)K);
#endif

  v8f c0 = {};
  v8f c1 = {};
  const float* bl0 = &Bl[m * BSTRIDE + 2 * half];
  const float* bl1 = &Bl[(m + 16) * BSTRIDE + 2 * half];
  const int nch = K / KC;

  for (int i = 0; i < nch; ++i) {
    const int kc = i * KC;
#if HAVE_TDM
    __builtin_amdgcn_s_wait_tensorcnt(0);   // wave0: chunk i landed (others: nop)
    __syncthreads();                        // publish chunk i (and Bl on i==0)
    if (wave == 0 && i + 1 < nch)           // prefetch chunk i+1 into other buf
      tdm_load_tile_f32(lds_addr_of(&Al[(i + 1) & 1][0]),
                        A + (size_t)rowBase * K + kc + KC,
                        KC, 128, (unsigned)(K - kc - KC),
                        (unsigned)(M - rowBase), (unsigned)K);
    const float* al0 = &Al[i & 1][(wave * 16 + m) * ASTRIDE + 2 * half];
#else
    __syncthreads();                        // Al reuse safe / Bl ready
    for (int v = tid; v < (128 * KC) / 4; v += 256) {
      int row = v >> 4;
      int kk  = (v & 15) * 4;
      int gr  = rowBase + row;
      float4 val = make_float4(0.f, 0.f, 0.f, 0.f);
      if (gr < M) val = *(const float4*)(A + (size_t)gr * K + kc + kk);
      *(float4*)&Al[0][row * ASTRIDE + kk] = val;
    }
    __syncthreads();
    const float* al0 = &Al[0][(wave * 16 + m) * ASTRIDE + 2 * half];
#endif

#pragma unroll
    for (int k = 0; k < KC; k += 4) {
      v2f a  = *(const v2f*)(al0 + k);            // A[row][kc+k+2h .. +1]
      v2f b0 = *(const v2f*)(bl0 + kc + k);       // B[kc+k+2h][col] pair
      v2f b1 = *(const v2f*)(bl1 + kc + k);
      c0 = __builtin_amdgcn_wmma_f32_16x16x4_f32(false, a, false, b0,
                                                 (short)0, c0, false, false);
      c1 = __builtin_amdgcn_wmma_f32_16x16x4_f32(false, a, false, b1,
                                                 (short)0, c1, false, false);
    }
#if HAVE_TDM
    __syncthreads();                        // all reads of chunk i done before
                                            // next TDM overwrites this buffer
#endif
  }

  if (row0 < M) {                       // wave-uniform (M % 16 == 0)
    float bb0 = bias ? bias[col0 + m]      : 0.0f;
    float bb1 = bias ? bias[col0 + 16 + m] : 0.0f;
#pragma unroll
    for (int r = 0; r < 8; ++r) {
      int row = row0 + r + 8 * half;    // C/D layout: vgpr r, lane half
      C[(size_t)row * Nc + col0 + m]      = c0[r] + bb0;
      C[(size_t)row * Nc + col0 + 16 + m] = c1[r] + bb1;
    }
  }
}

// ---------------- per-(node,head) attention logits --------------------------
__global__ void attn_logits(const float* __restrict__ Hb,
                            const float* __restrict__ a_src,
                            const float* __restrict__ a_dst,
                            float* __restrict__ als,
                            float* __restrict__ ald) {
  int idx = blockIdx.x * blockDim.x + threadIdx.x;   // n*NHEAD + h
  if (idx >= N_NODES * NHEAD) return;
  int h = idx & (NHEAD - 1);
  int n = idx >> 2;
  const float4* hv = (const float4*)(Hb + (size_t)n * FDIM + h * CH);
  const float4* as = (const float4*)(a_src + h * CH);
  const float4* ad = (const float4*)(a_dst + h * CH);
  float s = 0.f, d = 0.f;
#pragma unroll
  for (int i = 0; i < CH / 4; ++i) {
    float4 hh = hv[i], aa = as[i], dd = ad[i];
    s += hh.x * aa.x + hh.y * aa.y + hh.z * aa.z + hh.w * aa.w;
    d += hh.x * dd.x + hh.y * dd.y + hh.z * dd.z + hh.w * dd.w;
  }
  als[idx] = s;
  ald[idx] = d;
}

// ---------------- init / zero ----------------------------------------------
__global__ void init_seg(unsigned* __restrict__ emax, float* __restrict__ denom) {
  int idx = blockIdx.x * blockDim.x + threadIdx.x;
  if (idx < N_NODES * NHEAD) {
    emax[idx]  = fkey(-__builtin_inff());
    denom[idx] = 0.0f;
  }
}
__global__ void zero_f4(float4* __restrict__ p, int n4) {
  int i = blockIdx.x * blockDim.x + threadIdx.x;
  if (i < n4) p[i] = make_float4(0.f, 0.f, 0.f, 0.f);
}

// ---------------- edge helpers ----------------------------------------------
__device__ __forceinline__ void edge_sd(const int* __restrict__ ei, int e,
                                        int& s, int& d) {
  if (e < N_EDGES) { s = ei[e]; d = ei[N_EDGES + e]; }
  else             { s = e - N_EDGES; d = s; }       // self loops
}
__device__ __forceinline__ float leaky(float v) {
  return v > 0.f ? v : NEG_SLOPE * v;
}

// pass 1: segment max of leaky_relu logits
__global__ void edge_max(const int* __restrict__ ei,
                         const float* __restrict__ als,
                         const float* __restrict__ ald,
                         unsigned* __restrict__ emax) {
  int idx = blockIdx.x * blockDim.x + threadIdx.x;   // e*NHEAD + h
  const int TOT = (N_EDGES + N_NODES) * NHEAD;
  if (idx >= TOT) return;
  int h = idx & (NHEAD - 1), e = idx >> 2, s, d;
  edge_sd(ei, e, s, d);
  float v = leaky(als[s * NHEAD + h] + ald[d * NHEAD + h]);
  atomicMax(&emax[d * NHEAD + h], fkey(v));
}

// pass 2: softmax denominator
__global__ void edge_sum(const int* __restrict__ ei,
                         const float* __restrict__ als,
                         const float* __restrict__ ald,
                         const unsigned* __restrict__ emax,
                         float* __restrict__ denom) {
  int idx = blockIdx.x * blockDim.x + threadIdx.x;
  const int TOT = (N_EDGES + N_NODES) * NHEAD;
  if (idx >= TOT) return;
  int h = idx & (NHEAD - 1), e = idx >> 2, s, d;
  edge_sd(ei, e, s, d);
  float v = leaky(als[s * NHEAD + h] + ald[d * NHEAD + h]);
  atomicAdd(&denom[d * NHEAD + h], __expf(v - funkey(emax[d * NHEAD + h])));
}

// pass 3: alpha-weighted scatter-add. wave32 per edge; each lane owns one
// head-aligned 8-channel slab (lane>>3 = head, (lane&7)*8 = channel base).
__global__ void edge_aggr(const int* __restrict__ ei,
                          const float* __restrict__ als,
                          const float* __restrict__ ald,
                          const unsigned* __restrict__ emax,
                          const float* __restrict__ denom,
                          const float* __restrict__ Hb,
                          float* __restrict__ agg) {
  int gid  = blockIdx.x * blockDim.x + threadIdx.x;
  int e    = gid >> 5;
  int lane = gid & 31;
  const int TOT = N_EDGES + N_NODES;
  if (e >= TOT) return;                              // wave-uniform
  int s, d;
  edge_sd(ei, e, s, d);
  int h    = lane >> 3;
  int base = h * CH + (lane & 7) * 8;
  float v     = leaky(als[s * NHEAD + h] + ald[d * NHEAD + h]);
  float alpha = __expf(v - funkey(emax[d * NHEAD + h])) / denom[d * NHEAD + h];
  const float4* hp = (const float4*)(Hb + (size_t)s * FDIM + base);
  float4 h0 = hp[0], h1 = hp[1];
  float* op = agg + (size_t)d * FDIM + base;
  atomicAdd(op + 0, alpha * h0.x); atomicAdd(op + 1, alpha * h0.y);
  atomicAdd(op + 2, alpha * h0.z); atomicAdd(op + 3, alpha * h0.w);
  atomicAdd(op + 4, alpha * h1.x); atomicAdd(op + 5, alpha * h1.y);
  atomicAdd(op + 6, alpha * h1.z); atomicAdd(op + 7, alpha * h1.w);
}

// ---------------- bias + ELU (in place) -------------------------------------
__global__ void bias_elu(float* __restrict__ p, const float* __restrict__ b) {
  int i = blockIdx.x * blockDim.x + threadIdx.x;
  if (i >= N_NODES * FDIM) return;
  float v = p[i] + b[i & (FDIM - 1)];
  p[i] = v > 0.f ? v : expm1f(v);
}

// ---------------------------------------------------------------------------
extern "C" void kernel_launch(void* const* d_in, const int* in_sizes, int n_in,
                              void* d_out, int out_size, void* d_ws, size_t ws_size,
                              hipStream_t stream) {
  const float* x   = (const float*)d_in[0];
  const int*   ei  = (const int*)d_in[1];
  const float* W1  = (const float*)d_in[3];
  const float* as1 = (const float*)d_in[4];
  const float* ad1 = (const float*)d_in[5];
  const float* b1  = (const float*)d_in[6];
  const float* W2  = (const float*)d_in[7];
  const float* as2 = (const float*)d_in[8];
  const float* ad2 = (const float*)d_in[9];
  const float* b2  = (const float*)d_in[10];
  const float* fcW = (const float*)d_in[11];
  const float* fcb = (const float*)d_in[12];
  float* out = (float*)d_out;

  float*    bufA  = (float*)d_ws;                       // [N,256] gemm out / msgs
  float*    bufB  = bufA + (size_t)N_NODES * FDIM;      // [N,256] aggregation
  float*    als   = bufB + (size_t)N_NODES * FDIM;      // [N,H]
  float*    ald   = als + N_NODES * NHEAD;              // [N,H]
  unsigned* emax  = (unsigned*)(ald + N_NODES * NHEAD); // [N,H]
  float*    denom = (float*)(emax + N_NODES * NHEAD);   // [N,H]

  const int TPB   = 256;
  const int NH    = N_NODES * NHEAD;
  const int TOTE  = N_EDGES + N_NODES;
  const dim3 gGemm((N_NODES + 127) / 128, FDIM / 32);
  const dim3 gGemmFC((N_NODES + 127) / 128, OUTD / 32);
  const int gNH   = (NH + TPB - 1) / TPB;
  const int gEH   = (TOTE * NHEAD + TPB - 1) / TPB;
  const int gAgg  = (TOTE * 32 + TPB - 1) / TPB;
  const int gZero = (N_NODES * FDIM / 4 + TPB - 1) / TPB;
  const int gEW   = (N_NODES * FDIM + TPB - 1) / TPB;

  // ---- layer 1 ----
  gemm_wmma_f32<<<gGemm, TPB, 0, stream>>>(x, W1, bufA, nullptr,
                                           N_NODES, FDIM, FDIM);
  attn_logits<<<gNH, TPB, 0, stream>>>(bufA, as1, ad1, als, ald);
  init_seg<<<gNH, TPB, 0, stream>>>(emax, denom);
  zero_f4<<<gZero, TPB, 0, stream>>>((float4*)bufB, N_NODES * FDIM / 4);
  edge_max<<<gEH, TPB, 0, stream>>>(ei, als, ald, emax);
  edge_sum<<<gEH, TPB, 0, stream>>>(ei, als, ald, emax, denom);
  edge_aggr<<<gAgg, TPB, 0, stream>>>(ei, als, ald, emax, denom, bufA, bufB);
  bias_elu<<<gEW, TPB, 0, stream>>>(bufB, b1);

  // ---- layer 2 ----
  gemm_wmma_f32<<<gGemm, TPB, 0, stream>>>(bufB, W2, bufA, nullptr,
                                           N_NODES, FDIM, FDIM);
  attn_logits<<<gNH, TPB, 0, stream>>>(bufA, as2, ad2, als, ald);
  init_seg<<<gNH, TPB, 0, stream>>>(emax, denom);
  zero_f4<<<gZero, TPB, 0, stream>>>((float4*)bufB, N_NODES * FDIM / 4);
  edge_max<<<gEH, TPB, 0, stream>>>(ei, als, ald, emax);
  edge_sum<<<gEH, TPB, 0, stream>>>(ei, als, ald, emax, denom);
  edge_aggr<<<gAgg, TPB, 0, stream>>>(ei, als, ald, emax, denom, bufA, bufB);
  bias_elu<<<gEW, TPB, 0, stream>>>(bufB, b2);

  // ---- final FC (bias fused in gemm epilogue) ----
  gemm_wmma_f32<<<gGemmFC, TPB, 0, stream>>>(bufB, fcW, out, fcb,
                                             N_NODES, FDIM, OUTD);
}